// HistogramPatchAggregator_53480932769828
// MI455X (gfx1250) — compile-verified
//
#include <hip/hip_runtime.h>
#include <cstdint>
#include <cstddef>

// ---------------- problem constants ----------------
#define K7      7
#define PADR    3
#define IMG_H   512
#define IMG_W   512
#define TILE_W  32
#define TILE_H  16
#define HALO_W  (TILE_W + K7 - 1)   // 38
#define HALO_H  (TILE_H + K7 - 1)   // 22
#define NPIX    (HALO_W * HALO_H)   // 836 halo pixels
#define PXS     12                  // floats per pixel slot: c0..c2, g0..g6, s, pad
#define SMEM_F  (NPIX * PXS)        // 10032 floats = 40128 B LDS

#define INV_SC2 100.0f   // 1/0.1^2
#define INV_SG2 25.0f    // 1/0.2^2

#define AS1 __attribute__((address_space(1)))
#define AS3 __attribute__((address_space(3)))

// ---------------- CDNA5 async global->LDS path ----------------
#if defined(__has_builtin)
#  if __has_builtin(__builtin_amdgcn_global_load_async_to_lds_b32)
#    define HAVE_ASYNC_BUILTIN 1
#  endif
#  if __has_builtin(__builtin_amdgcn_s_wait_asynccnt)
#    define HAVE_WAITA_BUILTIN 1
#  endif
#endif

__device__ __forceinline__ void g2l_async_b32(const float* g, float* l) {
#if defined(HAVE_ASYNC_BUILTIN)
  // Builtin prototype uses int* operands: (global int*, lds int*, imm off, imm cpol)
  __builtin_amdgcn_global_load_async_to_lds_b32(
      (AS1 int*)(int*)(void*)const_cast<float*>(g),
      (AS3 int*)(int*)(void*)l, 0, 0);
#else
  unsigned int loff = (unsigned int)(uintptr_t)(AS3 void*)(void*)l;
  asm volatile("global_load_async_to_lds_b32 %0, %1, off"
               :: "v"(loff), "v"(g) : "memory");
#endif
}

__device__ __forceinline__ void wait_async_zero() {
#if defined(HAVE_WAITA_BUILTIN)
  __builtin_amdgcn_s_wait_asynccnt(0);
#else
  asm volatile("s_wait_asynccnt 0x0" ::: "memory");
#endif
}

// ---------------- kernel ----------------
__global__ void __launch_bounds__(256)
jbf7x7_gfx1250(const float* __restrict__ img,   // [B,3,H,W]
               const float* __restrict__ gui,   // [B,7,H,W]
               float* __restrict__ out) {       // [B,3,H,W]
  __shared__ __align__(16) float smem[SMEM_F];

  const int tid = threadIdx.x;
  const int tx  = tid & 31;        // 0..31
  const int ty  = tid >> 5;        // 0..7
  const int gx0 = blockIdx.x * TILE_W;
  const int gy0 = blockIdx.y * TILE_H;
  const int b   = blockIdx.z;
  const size_t HW = (size_t)IMG_H * IMG_W;

  // Phase 0: zero LDS tile (establishes zero padding for out-of-image halo).
  {
    float4 z = make_float4(0.f, 0.f, 0.f, 0.f);
    float4* s4 = (float4*)smem;
    for (int i = tid; i < SMEM_F / 4; i += 256) s4[i] = z;
  }
  __syncthreads();

  // Phase 1: async-copy the in-bounds halo into LDS (per-lane scatter, b32).
  // One coordinate computation per halo pixel; 10 plane-strided async loads.
  {
    const float* ib = img + (size_t)(b * 3) * HW;
    const float* gb = gui + (size_t)(b * 7) * HW;
    for (int idx = tid; idx < NPIX; idx += 256) {
      const int py = idx / HALO_W;
      const int px = idx - py * HALO_W;
      const int gy = gy0 - PADR + py;
      const int gx = gx0 - PADR + px;
      if ((unsigned)gy < (unsigned)IMG_H && (unsigned)gx < (unsigned)IMG_W) {
        const size_t goff = (size_t)gy * IMG_W + gx;
        const float* ip = ib + goff;
        const float* gp = gb + goff;
        float* l = &smem[idx * PXS];
#pragma unroll
        for (int c = 0; c < 3; ++c)
          g2l_async_b32(ip + (size_t)c * HW, l + c);
#pragma unroll
        for (int c = 0; c < 7; ++c)
          g2l_async_b32(gp + (size_t)c * HW, l + 3 + c);
      }
    }
  }
  wait_async_zero();
  __syncthreads();

  // Phase 2: per halo pixel, precompute scaled squared norm s into slot 10:
  //   s = 100*||c||^2 + 25*||g||^2   (zeros for padded pixels -> s = 0)
  for (int idx = tid; idx < NPIX; idx += 256) {
    float* p = &smem[idx * PXS];
    const float4 a  = *(const float4*)(p);      // c0 c1 c2 g0
    const float4 bb = *(const float4*)(p + 4);  // g1 g2 g3 g4
    const float4 cc = *(const float4*)(p + 8);  // g5 g6 s pad
    float s = INV_SC2 * (a.x * a.x + a.y * a.y + a.z * a.z)
            + INV_SG2 * (a.w * a.w + bb.x * bb.x + bb.y * bb.y +
                         bb.z * bb.z + bb.w * bb.w + cc.x * cc.x + cc.y * cc.y);
    p[10] = s;
  }
  __syncthreads();

  // Phase 3: bilateral filtering. Each thread produces 2 pixels (rows ty, ty+8).
  for (int r = 0; r < 2; ++r) {
    const int ly = ty + r * 8;  // tile row 0..15

    // Center values, with distance scales folded in.
    const float* cp = &smem[((ly + PADR) * HALO_W + (tx + PADR)) * PXS];
    const float4 a  = *(const float4*)(cp);
    const float4 bq = *(const float4*)(cp + 4);
    const float4 cq = *(const float4*)(cp + 8);
    const float cs0 = INV_SC2 * a.x,  cs1 = INV_SC2 * a.y, cs2 = INV_SC2 * a.z;
    const float cs3 = INV_SG2 * a.w,  cs4 = INV_SG2 * bq.x, cs5 = INV_SG2 * bq.y;
    const float cs6 = INV_SG2 * bq.z, cs7 = INV_SG2 * bq.w;
    const float cs8 = INV_SG2 * cq.x, cs9 = INV_SG2 * cq.y;
    const float sc  = cq.z;

    float n0 = 0.f, n1 = 0.f, n2 = 0.f, den = 0.f;

    const float* tp = &smem[(ly * HALO_W + tx) * PXS];
#pragma unroll
    for (int dy = 0; dy < K7; ++dy) {
      const float* rp = tp + dy * (HALO_W * PXS);
#pragma unroll
      for (int dx = 0; dx < K7; ++dx) {
        const float* qp = rp + dx * PXS;
        const float4 q0 = *(const float4*)(qp);      // c0 c1 c2 g0
        const float4 q1 = *(const float4*)(qp + 4);  // g1 g2 g3 g4
        const float4 q2 = *(const float4*)(qp + 8);  // g5 g6 s  pad

        // E = s(p) + s(c) - 2 * sum_i w_i * p_i * c_i
        float dot = q0.x * cs0;
        dot = fmaf(q0.y, cs1, dot);
        dot = fmaf(q0.z, cs2, dot);
        dot = fmaf(q0.w, cs3, dot);
        dot = fmaf(q1.x, cs4, dot);
        dot = fmaf(q1.y, cs5, dot);
        dot = fmaf(q1.z, cs6, dot);
        dot = fmaf(q1.w, cs7, dot);
        dot = fmaf(q2.x, cs8, dot);
        dot = fmaf(q2.y, cs9, dot);
        const float E = fmaf(-2.0f, dot, sc + q2.z);
        const float w = __expf(-E);                 // v_exp_f32 (trans unit)

        n0  = fmaf(q0.x, w, n0);
        n1  = fmaf(q0.y, w, n1);
        n2  = fmaf(q0.z, w, n2);
        den += w;
      }
    }

    const int gy = gy0 + ly;
    const int gx = gx0 + tx;
    const float d = den + 1e-6f;
    const size_t o = ((size_t)(b * 3) * IMG_H + gy) * IMG_W + gx;
    out[o]          = n0 / d;
    out[o + HW]     = n1 / d;
    out[o + 2 * HW] = n2 / d;
  }
}

// ---------------- launch ----------------
extern "C" void kernel_launch(void* const* d_in, const int* in_sizes, int n_in,
                              void* d_out, int out_size, void* d_ws, size_t ws_size,
                              hipStream_t stream) {
  const float* img = (const float*)d_in[0];  // output  [B,3,512,512] f32
  const float* gui = (const float*)d_in[1];  // guidance[B,7,512,512] f32
  float* out = (float*)d_out;                // [B,3,512,512] f32

  const int B = in_sizes[0] / (3 * IMG_H * IMG_W);  // = 2
  dim3 grid(IMG_W / TILE_W, IMG_H / TILE_H, B);
  dim3 block(256);
  jbf7x7_gfx1250<<<grid, block, 0, stream>>>(img, gui, out);
}